// LAL_46858093199670
// MI455X (gfx1250) — compile-verified
//
#include <hip/hip_runtime.h>

// Problem constants from the reference: x:[4,4096,64] f32 -> out:[4,4096,4096] f32
#define B_ 4
#define N_ 4096
#define D_ 64

typedef __attribute__((ext_vector_type(2))) float v2f;
typedef __attribute__((ext_vector_type(8))) float v8f;

// Pass 0: per-row sum of squares (B*N values) + init global min/max (d^2 domain).
__global__ void sq_init_kernel(const float* __restrict__ x, float* __restrict__ sq,
                               unsigned int* __restrict__ mm) {
  const int idx = blockIdx.x * blockDim.x + threadIdx.x;
  if (idx == 0) {
    mm[0] = 0x7F800000u;  // +inf  (min slot, d^2 bits)
    mm[1] = 0u;           // 0.0f  (max slot; d^2 >= 0)
  }
  if (idx < B_ * N_) {
    const float* row = x + (size_t)idx * D_;
    float s = 0.0f;
#pragma unroll
    for (int k = 0; k < D_; k += 4) {
      float4 v = *(const float4*)(row + k);
      s = fmaf(v.x, v.x, s);
      s = fmaf(v.y, v.y, s);
      s = fmaf(v.z, v.z, s);
      s = fmaf(v.w, v.w, s);
    }
    sq[idx] = s;
  }
}

// Each wave computes a 32x32 output patch = 2x2 register tile of 16x16 WMMA tiles
// (64 chained V_WMMA_F32_16X16X4_F32, K=64, 1.0 fragment loads per WMMA).
// FINAL=false: reduce global min/max of clamped d^2 (sqrt deferred — monotone).
// FINAL=true : recompute (deterministic), sqrt, normalize, set diagonal, NT-store.
template <bool FINAL>
__global__ void cdist_kernel(const float* __restrict__ x, const float* __restrict__ sq,
                             float* __restrict__ out, unsigned int* __restrict__ mm) {
  const int b  = blockIdx.z;
  const int wv = threadIdx.x >> 5;   // 8 waves: 2 (rows) x 4 (cols)
  const int ln = threadIdx.x & 31;
  const int hi = ln >> 4;            // K half: lanes 0-15 -> K{0,1}, 16-31 -> K{2,3}
  const int m  = ln & 15;            // row (A) / col (B) within a 16-tile
  const int i0 = (blockIdx.y << 6) + ((wv >> 2) << 5);  // 32-row wave tile base
  const int j0 = (blockIdx.x << 7) + ((wv & 3) << 5);   // 32-col wave tile base

  // ISA 7.12.2 f32 16x4 / 4x16 fragment layout: lane holds K = 2*hi + {0,1}.
  const float* ra0 = x + (size_t)(b * N_ + i0 + m) * D_ + 2 * hi;
  const float* ra1 = ra0 + (size_t)16 * D_;
  const float* rb0 = x + (size_t)(b * N_ + j0 + m) * D_ + 2 * hi;
  const float* rb1 = rb0 + (size_t)16 * D_;

  v8f acc[2][2] = {{{}, {}}, {{}, {}}};
#pragma unroll
  for (int s = 0; s < D_ / 4; ++s) {
    v2f a0 = *(const v2f*)(ra0 + 4 * s);  // 8B-aligned float2 fragments
    v2f a1 = *(const v2f*)(ra1 + 4 * s);
    v2f b0 = *(const v2f*)(rb0 + 4 * s);
    v2f b1 = *(const v2f*)(rb1 + 4 * s);
    acc[0][0] = __builtin_amdgcn_wmma_f32_16x16x4_f32(false, a0, false, b0, (short)0,
                                                      acc[0][0], false, false);
    acc[0][1] = __builtin_amdgcn_wmma_f32_16x16x4_f32(false, a0, false, b1, (short)0,
                                                      acc[0][1], false, false);
    acc[1][0] = __builtin_amdgcn_wmma_f32_16x16x4_f32(false, a1, false, b0, (short)0,
                                                      acc[1][0], false, false);
    acc[1][1] = __builtin_amdgcn_wmma_f32_16x16x4_f32(false, a1, false, b1, (short)0,
                                                      acc[1][1], false, false);
  }

  const float* sqb = sq + b * N_;
  const float sqj[2] = {sqb[j0 + m], sqb[j0 + 16 + m]};  // C/D layout: N = m

  if (FINAL) {
    // min/max were reduced in the d^2 domain; sqrt is monotone so bounds carry over.
    const float dmin = sqrtf(__uint_as_float(mm[0]));
    const float dmax = sqrtf(__uint_as_float(mm[1]));
    const float inv  = 1.0f / (dmax - dmin);
#pragma unroll
    for (int ti = 0; ti < 2; ++ti) {
#pragma unroll
      for (int r = 0; r < 8; ++r) {
        const int   gi  = i0 + 16 * ti + r + 8 * hi;  // C/D layout: M = r + 8*hi
        const float sqi = sqb[gi];
        float* orow = out + (size_t)(b * N_ + gi) * N_;
#pragma unroll
        for (int tj = 0; tj < 2; ++tj) {
          const int gj = j0 + 16 * tj + m;
          float d2  = sqi + sqj[tj] - 2.0f * acc[ti][tj][r];
          float dd  = sqrtf(fmaxf(d2, 0.0f));
          float val = (dd - dmin) * inv;
          if (gi == gj) val = 1.0f;                       // diagonal := 1
          __builtin_nontemporal_store(val, orow + gj);    // stream 256MB past L2
        }
      }
    }
  } else {
    float lmin = 3.0e38f, lmax = 0.0f;
#pragma unroll
    for (int ti = 0; ti < 2; ++ti) {
#pragma unroll
      for (int r = 0; r < 8; ++r) {
        const float sqi = sqb[i0 + 16 * ti + r + 8 * hi];
#pragma unroll
        for (int tj = 0; tj < 2; ++tj) {
          float d2 = fmaxf(sqi + sqj[tj] - 2.0f * acc[ti][tj][r], 0.0f);
          lmin = fminf(lmin, d2);
          lmax = fmaxf(lmax, d2);
        }
      }
    }
    __shared__ float smin[256];
    __shared__ float smax[256];
    const int tid = threadIdx.x;
    smin[tid] = lmin;
    smax[tid] = lmax;
    __syncthreads();
    for (int s = 128; s > 0; s >>= 1) {
      if (tid < s) {
        smin[tid] = fminf(smin[tid], smin[tid + s]);
        smax[tid] = fmaxf(smax[tid], smax[tid + s]);
      }
      __syncthreads();
    }
    if (tid == 0) {
      // Nonnegative floats order like their uint bit patterns.
      atomicMin(&mm[0], __float_as_uint(smin[0]));
      atomicMax(&mm[1], __float_as_uint(smax[0]));
    }
  }
}

extern "C" void kernel_launch(void* const* d_in, const int* in_sizes, int n_in,
                              void* d_out, int out_size, void* d_ws, size_t ws_size,
                              hipStream_t stream) {
  const float* x   = (const float*)d_in[0];
  float*       out = (float*)d_out;

  // Workspace layout: [0..7]: uint min/max d^2 bits; 256B-aligned sq[B*N] floats (64 KB).
  unsigned int* mm = (unsigned int*)d_ws;
  float*        sq = (float*)((char*)d_ws + 256);

  sq_init_kernel<<<(B_ * N_ + 255) / 256, 256, 0, stream>>>(x, sq, mm);

  dim3 grid(N_ / 128, N_ / 64, B_);  // (32, 64, 4); 256 threads = 8 waves (2x4) per block
  cdist_kernel<false><<<grid, 256, 0, stream>>>(x, sq, out, mm);  // d^2 min/max (no big writes)
  cdist_kernel<true ><<<grid, 256, 0, stream>>>(x, sq, out, mm);  // recompute + normalize + store
}